// Net_6425271075083
// MI455X (gfx1250) — compile-verified
//
#include <hip/hip_runtime.h>

typedef __attribute__((ext_vector_type(2))) float v2f;
typedef __attribute__((ext_vector_type(8))) float v8f;
typedef __attribute__((__vector_size__(16))) int v4i_t;

#define N_USERS   10000
#define N_ITEMS   1000
#define N_RATINGS 5
#define E_PER_R   200000
#define IN_UNITS  256
#define AGG_UNITS 256
#define OUT_UNITS 64
#define HIST_LEN  200
#define LEAKY_SLOPE 0.1f

// LDS row stride for the staged 256x64 W panel: 64 + 8 pad floats.
// word bank = (72*k + c) mod 64 = (8k + c) mod 64 -> the two half-wave groups
// (k0 vs k0+2) land on disjoint 16-bank sets: conflict-free ds_load_b32.
#define LDS_STRIDE 72

#if defined(__has_builtin)
#if __has_builtin(__builtin_amdgcn_global_load_async_to_lds_b128)
#define HAVE_ASYNC_LDS 1
#endif
#endif

__device__ __forceinline__ float leaky(float x) {
    return x > 0.0f ? x : LEAKY_SLOPE * x;
}

// ---------------------------------------------------------------------------
// Degree histogram over all edges (union of the 5 rating graphs)
// ---------------------------------------------------------------------------
__global__ void degree_kernel(const int* __restrict__ src, const int* __restrict__ dst,
                              float* __restrict__ deg_u, float* __restrict__ deg_i, int nE) {
    int i = blockIdx.x * blockDim.x + threadIdx.x;
    if (i < nE) {
        atomicAdd(&deg_u[src[i]], 1.0f);
        atomicAdd(&deg_i[dst[i]], 1.0f);
    }
}

__global__ void inv_sqrt_kernel(float* __restrict__ d, int n) {
    int i = blockIdx.x * blockDim.x + threadIdx.x;
    if (i < n) d[i] = rsqrtf(fmaxf(d[i], 1.0f));
}

// ---------------------------------------------------------------------------
// H[M,N] = act( X[M,256] * c[row] ) @ W[256,N] (+ bias[col])
//   act==0 : plain (message GEMM), act==1 : LeakyReLU(0.1)+bias (fc GEMM)
// Block = 256 threads = 8 waves. Each block owns a 64-col n-group; the
// 256x64 W panel is staged once into LDS (async-to-LDS when available),
// then each wave computes a 16x64 strip with V_WMMA_F32_16X16X4_F32.
// A-frag: lane row = lane&15, k = 2*(lane>>4)+j (contiguous pair -> b64 load)
// B-frag: lane col = lane&15, k = 2*(lane>>4)+j (two ds_load_b32 from LDS)
// D:      lane (M,N) = (v + 8*(lane>>4), lane&15)
// ---------------------------------------------------------------------------
__global__ __launch_bounds__(256)
void gemm_rowscale_wmma(const float* __restrict__ X, const float* __restrict__ c,
                        const float* __restrict__ W, const float* __restrict__ bias,
                        float* __restrict__ H, int M, int N, int act, int mtiles) {
    __shared__ float wpanel[256 * LDS_STRIDE];   // 72 KB of the 320 KB WGP LDS

    const int t  = threadIdx.x;
    const int n0 = blockIdx.y << 6;

    // ---- stage W[k][n0..n0+63] -> wpanel[k*72 + c], 16B chunks ----
#pragma unroll
    for (int i = 0; i < 16; ++i) {
        int idx = t + (i << 8);                  // 0..4095 chunk id
        int k   = idx >> 4;                      // 0..255
        int cc  = (idx & 15) << 2;               // 0,4,...,60
        const float* gsrc = W + (size_t)k * N + n0 + cc;
        float*       ldst = &wpanel[k * LDS_STRIDE + cc];
#if HAVE_ASYNC_LDS
        __builtin_amdgcn_global_load_async_to_lds_b128(
            (__attribute__((address_space(1))) v4i_t*)(float*)gsrc,
            (__attribute__((address_space(3))) v4i_t*)ldst, 0, 0);
#else
        *(float4*)ldst = *(const float4*)gsrc;
#endif
    }
#if HAVE_ASYNC_LDS
    asm volatile("s_wait_asynccnt 0" ::: "memory");
#endif
    __syncthreads();

    // ---- per-wave 16x64 WMMA strip ----
    const int wave = t >> 5;
    const int mt   = blockIdx.x * 8 + wave;
    if (mt >= mtiles) return;                    // uniform per wave, after barrier

    const int lane = t & 31;
    const int lr   = lane & 15;
    const int kh   = lane >> 4;
    const int m0   = mt << 4;

    const int arow = m0 + lr;
    float scale = 0.0f;
    const float* Xrow = X;
    if (arow < M) { scale = c[arow]; Xrow = X + (size_t)arow * 256; }

    v8f acc0 = {}, acc1 = {}, acc2 = {}, acc3 = {};
#pragma unroll 2
    for (int kk = 0; kk < 256; kk += 4) {
        int k0 = kk + (kh << 1);
        v2f a = *(const v2f*)(Xrow + k0);        // contiguous pair -> b64
        a.x *= scale; a.y *= scale;
        if (act) { a.x = leaky(a.x); a.y = leaky(a.y); }

        const float* l0 = &wpanel[k0 * LDS_STRIDE + lr];
        const float* l1 = l0 + LDS_STRIDE;
        v2f b;
        b.x = l0[0];  b.y = l1[0];
        acc0 = __builtin_amdgcn_wmma_f32_16x16x4_f32(false, a, false, b, (short)0, acc0, false, false);
        b.x = l0[16]; b.y = l1[16];
        acc1 = __builtin_amdgcn_wmma_f32_16x16x4_f32(false, a, false, b, (short)0, acc1, false, false);
        b.x = l0[32]; b.y = l1[32];
        acc2 = __builtin_amdgcn_wmma_f32_16x16x4_f32(false, a, false, b, (short)0, acc2, false, false);
        b.x = l0[48]; b.y = l1[48];
        acc3 = __builtin_amdgcn_wmma_f32_16x16x4_f32(false, a, false, b, (short)0, acc3, false, false);
    }

    float bv0 = 0.f, bv1 = 0.f, bv2 = 0.f, bv3 = 0.f;
    if (bias) {
        bv0 = bias[n0 + lr];      bv1 = bias[n0 + 16 + lr];
        bv2 = bias[n0 + 32 + lr]; bv3 = bias[n0 + 48 + lr];
    }
#pragma unroll
    for (int v = 0; v < 8; ++v) {
        int r = m0 + v + (kh << 3);
        if (r < M) {
            float* Hr = H + (size_t)r * N + n0 + lr;
            Hr[0]  = acc0[v] + bv0;
            Hr[16] = acc1[v] + bv1;
            Hr[32] = acc2[v] + bv2;
            Hr[48] = acc3[v] + bv3;
        }
    }
}

// ---------------------------------------------------------------------------
// Per-edge message scatter: item_agg[dst] += hu[src]; user_agg[src] += hi[dst]
// One block (256 lanes) per edge; L2-resident float atomics.
// ---------------------------------------------------------------------------
__global__ void scatter_kernel(const float* __restrict__ hu, const float* __restrict__ hi,
                               const int* __restrict__ src, const int* __restrict__ dst,
                               float* __restrict__ user_agg, float* __restrict__ item_agg) {
    int e = blockIdx.x;
    int t = threadIdx.x;
    int s = src[e];
    int d = dst[e];
    atomicAdd(&item_agg[(size_t)d * AGG_UNITS + t], hu[(size_t)s * AGG_UNITS + t]);
    atomicAdd(&user_agg[(size_t)s * AGG_UNITS + t], hi[(size_t)d * AGG_UNITS + t]);
}

// ---------------------------------------------------------------------------
// p[u] += sum_h Y0[imp[u,h]] / sqrt_count[u]   (Y row 0 treated as zero)
// ---------------------------------------------------------------------------
__global__ void implicit_add_kernel(const int* __restrict__ imp, const float* __restrict__ sc,
                                    const float* __restrict__ Y, float* __restrict__ P) {
    int u = blockIdx.x;
    int t = threadIdx.x;                 // 64 threads, one per OUT dim
    const int* row = imp + (size_t)u * HIST_LEN;
    float acc = 0.0f;
    for (int h = 0; h < HIST_LEN; ++h) {
        int idx = row[h];
        if (idx != 0) acc += Y[(size_t)idx * OUT_UNITS + t];
    }
    P[(size_t)u * OUT_UNITS + t] += acc / sc[u];
}

// ---------------------------------------------------------------------------
// out[i,u] = dot(q[i,:], py[u,:]) + bi[i] + bu[u] + gmean
// M=1000 (items) x N=10000 (users), K=64. One wave -> 16x64 strip.
// All fragment loads are contiguous f32 pairs -> global_load_b64.
// ---------------------------------------------------------------------------
__global__ __launch_bounds__(128)
void final_wmma_kernel(const float* __restrict__ Q, const float* __restrict__ PY,
                       const float* __restrict__ bi, const float* __restrict__ bu,
                       const float* __restrict__ gmean, float* __restrict__ out,
                       int nwaves) {
    const int M = N_ITEMS, N = N_USERS, K = OUT_UNITS;
    const int NTILES  = N / 16;               // 625
    const int NGROUPS = (NTILES + 3) / 4;     // 157

    int wave = blockIdx.x * (blockDim.x >> 5) + (threadIdx.x >> 5);
    if (wave >= nwaves) return;
    int mt = wave / NGROUPS;
    int ng = wave - mt * NGROUPS;

    int lane = threadIdx.x & 31;
    int lr   = lane & 15;
    int kh   = lane >> 4;
    int m0   = mt << 4;
    int ntb  = ng << 2;

    int arow = m0 + lr;
    float amask = 1.0f;
    if (arow >= M) { arow = M - 1; amask = 0.0f; }
    const float* Qrow = Q + (size_t)arow * K;

    int nt1 = min(ntb + 1, NTILES - 1);
    int nt2 = min(ntb + 2, NTILES - 1);
    int nt3 = min(ntb + 3, NTILES - 1);
    const float* p0 = PY + (size_t)(ntb * 16 + lr) * K;
    const float* p1 = PY + (size_t)(nt1 * 16 + lr) * K;
    const float* p2 = PY + (size_t)(nt2 * 16 + lr) * K;
    const float* p3 = PY + (size_t)(nt3 * 16 + lr) * K;

    v8f acc0 = {}, acc1 = {}, acc2 = {}, acc3 = {};
#pragma unroll 4
    for (int kk = 0; kk < K; kk += 4) {
        int k0 = kk + (kh << 1);
        v2f a = *(const v2f*)(Qrow + k0);
        a.x *= amask; a.y *= amask;
        v2f b;
        b = *(const v2f*)(p0 + k0);
        acc0 = __builtin_amdgcn_wmma_f32_16x16x4_f32(false, a, false, b, (short)0, acc0, false, false);
        b = *(const v2f*)(p1 + k0);
        acc1 = __builtin_amdgcn_wmma_f32_16x16x4_f32(false, a, false, b, (short)0, acc1, false, false);
        b = *(const v2f*)(p2 + k0);
        acc2 = __builtin_amdgcn_wmma_f32_16x16x4_f32(false, a, false, b, (short)0, acc2, false, false);
        b = *(const v2f*)(p3 + k0);
        acc3 = __builtin_amdgcn_wmma_f32_16x16x4_f32(false, a, false, b, (short)0, acc3, false, false);
    }

    float gm = gmean[0];
#pragma unroll
    for (int j = 0; j < 4; ++j) {
        int nt = ntb + j;
        if (nt >= NTILES) break;
        v8f acc = (j == 0) ? acc0 : (j == 1) ? acc1 : (j == 2) ? acc2 : acc3;
        int col = nt * 16 + lr;
        float badd = bu[col] + gm;
#pragma unroll
        for (int v = 0; v < 8; ++v) {
            int r = m0 + v + (kh << 3);
            if (r < M) out[(size_t)r * N + col] = acc[v] + bi[r] + badd;
        }
    }
}

// ---------------------------------------------------------------------------
// Orchestration
// ---------------------------------------------------------------------------
extern "C" void kernel_launch(void* const* d_in, const int* in_sizes, int n_in,
                              void* d_out, int out_size, void* d_ws, size_t ws_size,
                              hipStream_t stream) {
    const int*   src        = (const int*)d_in[0];   // [5, 200000]
    const int*   dst        = (const int*)d_in[1];   // [5, 200000]
    const int*   imp        = (const int*)d_in[2];   // [10000, 200]
    const float* sqrt_count = (const float*)d_in[3]; // [10000, 1]
    const float* gmean      = (const float*)d_in[4]; // [1]
    const float* ufeat      = (const float*)d_in[5]; // [10000, 256]
    const float* ifeat      = (const float*)d_in[6]; // [1000, 256]
    const float* W          = (const float*)d_in[7]; // [5, 256, 256]
    const float* fc_w       = (const float*)d_in[8]; // [256, 64]
    const float* fc_b       = (const float*)d_in[9]; // [64]
    const float* bu         = (const float*)d_in[10];// [10000, 1]
    const float* bi         = (const float*)d_in[11];// [1000, 1]
    const float* Y          = (const float*)d_in[12];// [1001, 64]
    float*       out        = (float*)d_out;         // [1000, 10000]

    // Workspace layout (floats), ~25.4 MB total
    float* ws   = (float*)d_ws;
    float* cu   = ws;                                       // 10016 (deg_u -> rsqrt)
    float* ci   = cu + 10016;                               // 1008  (deg_i -> rsqrt)
    float* hu   = ci + 1008;                                // 10000*256
    float* hi   = hu + (size_t)N_USERS * AGG_UNITS;         // 1000*256
    float* uagg = hi + (size_t)N_ITEMS * AGG_UNITS;         // 10000*256
    float* iagg = uagg + (size_t)N_USERS * AGG_UNITS;       // 1000*256
    float* p    = iagg + (size_t)N_ITEMS * AGG_UNITS;       // 10000*64
    float* q    = p + (size_t)N_USERS * OUT_UNITS;          // 1000*64

    // Zero the accumulation buffers (graph-capturable async memset)
    (void)hipMemsetAsync(cu, 0, (10016 + 1008) * sizeof(float), stream);
    (void)hipMemsetAsync(uagg, 0, ((size_t)N_USERS + N_ITEMS) * AGG_UNITS * sizeof(float), stream);

    // 1) degrees -> symmetric normalization coefficients
    int nE = N_RATINGS * E_PER_R;
    degree_kernel<<<(nE + 255) / 256, 256, 0, stream>>>(src, dst, cu, ci, nE);
    inv_sqrt_kernel<<<(N_USERS + 255) / 256, 256, 0, stream>>>(cu, N_USERS);
    inv_sqrt_kernel<<<(N_ITEMS + 255) / 256, 256, 0, stream>>>(ci, N_ITEMS);

    // 2) per-rating message GEMMs + edge scatter
    const int MT_U = N_USERS / 16;          // 625
    const int MT_I = (N_ITEMS + 15) / 16;   // 63
    const int MB_U = (MT_U + 7) / 8;        // 79 blocks of 8 waves
    const int MB_I = (MT_I + 7) / 8;        // 8
    for (int r = 0; r < N_RATINGS; ++r) {
        const float* Wr = W + (size_t)r * IN_UNITS * AGG_UNITS;
        gemm_rowscale_wmma<<<dim3(MB_U, AGG_UNITS / 64), 256, 0, stream>>>(
            ufeat, cu, Wr, nullptr, hu, N_USERS, AGG_UNITS, 0, MT_U);
        gemm_rowscale_wmma<<<dim3(MB_I, AGG_UNITS / 64), 256, 0, stream>>>(
            ifeat, ci, Wr, nullptr, hi, N_ITEMS, AGG_UNITS, 0, MT_I);
        scatter_kernel<<<E_PER_R, AGG_UNITS, 0, stream>>>(
            hu, hi, src + (size_t)r * E_PER_R, dst + (size_t)r * E_PER_R, uagg, iagg);
    }

    // 3) fc layer: p = leaky(uagg*cu) @ fc_w + fc_b ; q likewise
    gemm_rowscale_wmma<<<dim3(MB_U, 1), 256, 0, stream>>>(
        uagg, cu, fc_w, fc_b, p, N_USERS, OUT_UNITS, 1, MT_U);
    gemm_rowscale_wmma<<<dim3(MB_I, 1), 256, 0, stream>>>(
        iagg, ci, fc_w, fc_b, q, N_ITEMS, OUT_UNITS, 1, MT_I);

    // 4) implicit-feedback pooling added into p
    implicit_add_kernel<<<N_USERS, OUT_UNITS, 0, stream>>>(imp, sqrt_count, Y, p);

    // 5) out = q @ (p+y)^T + bi + bu^T + gmean
    int wf = MT_I * ((N_USERS / 16 + 3) / 4);   // 63 * 157
    final_wmma_kernel<<<(wf + 3) / 4, 128, 0, stream>>>(q, p, bi, bu, gmean, out, wf);
}